// ViterbiDecoder_36240934044213
// MI455X (gfx1250) — compile-verified
//
#include <hip/hip_runtime.h>
#include <hip/hip_bf16.h>
#include <stdint.h>

// Problem constants (match reference: B=64, T=2048, S=256)
#define BATCH    64
#define TLEN     2048
#define NSTATE   256
#define NTHREADS 256   // one thread per destination state j; 8 waves (wave32)
#define NCHAIN   8     // independent argmax accumulator chains (ILP)

// ---------------------------------------------------------------------------
// CDNA5 async global->LDS path (gfx1250), signatures verified by compile:
//   __builtin_amdgcn_global_load_async_to_lds_b32 (AS1 int*  src, AS3 int*  dst, Ii off, Ii cpol)
//   __builtin_amdgcn_global_load_async_to_lds_b128(AS1 int4* src, AS3 int4* dst, Ii off, Ii cpol)
// ---------------------------------------------------------------------------
#if defined(__has_builtin)
#if __has_builtin(__builtin_amdgcn_global_load_async_to_lds_b32) && \
    __has_builtin(__builtin_amdgcn_global_load_async_to_lds_b128)
#define HAVE_ASYNC_LDS 1
#endif
#endif
#ifndef HAVE_ASYNC_LDS
#define HAVE_ASYNC_LDS 0
#endif

#if HAVE_ASYNC_LDS
typedef int v4i __attribute__((ext_vector_type(4)));
typedef __attribute__((address_space(1))) int  as1_i32;
typedef __attribute__((address_space(3))) int  as3_i32;
typedef __attribute__((address_space(1))) v4i  as1_v4i;
typedef __attribute__((address_space(3))) v4i  as3_v4i;

__device__ __forceinline__ void async_load_b32(const void* g, void* l) {
  // Generic LDS pointers carry the LDS offset in the low 32 bits on AMDGPU.
  __builtin_amdgcn_global_load_async_to_lds_b32(
      (as1_i32*)(uintptr_t)g, (as3_i32*)(uint32_t)(uintptr_t)l, 0, 0);
}
__device__ __forceinline__ void async_load_b128(const void* g, void* l) {
  __builtin_amdgcn_global_load_async_to_lds_b128(
      (as1_v4i*)(uintptr_t)g, (as3_v4i*)(uint32_t)(uintptr_t)l, 0, 0);
}
__device__ __forceinline__ void wait_async_le0() {
  asm volatile("s_wait_asynccnt 0x0" ::: "memory");
}
__device__ __forceinline__ void wait_async_le1() {
  asm volatile("s_wait_asynccnt 0x1" ::: "memory");
}
#endif

// ---------------------------------------------------------------------------
// Forward Viterbi: one workgroup per batch element. log_trans lives entirely
// in LDS (256KB -- needs CDNA5's 320KB WGP LDS). delta ping-pongs in LDS.
// Emissions are double-buffered in LDS via async loads (each lane fetches its
// own element, so a per-wave s_wait_asynccnt suffices -- no barrier needed).
// Backpointers are uint8 (S=256) streamed to the workspace.
// ---------------------------------------------------------------------------
__global__ __launch_bounds__(NTHREADS, 1)
void viterbi_fwd_kernel(const float* __restrict__ em,     // [B,T,S]
                        const float* __restrict__ lt,     // [S,S] prev->next
                        uint8_t* __restrict__ bp,         // [T-1,B,S]
                        int* __restrict__ last_state,     // [B]
                        float* __restrict__ out_base)     // d_out as float*
{
  extern __shared__ float smem[];
  float* tr  = smem;                          // 65536 floats (256 KB)
  float* dA  = tr + NSTATE * NSTATE;          // 256 floats
  float* dB  = dA + NSTATE;                   // 256 floats
  float* emb = dB + NSTATE;                   // 2 * 256 floats (double buffer)

  const int j = threadIdx.x;                  // destination state
  const int b = blockIdx.x;                   // batch element
  const float* emG = em + (size_t)b * TLEN * NSTATE;

  // ---- Stage log_trans into LDS (row major: tr[i*S + j]) -------------------
  // Uniform trip count (64) so no divergent exec-masked loop is generated.
#if HAVE_ASYNC_LDS
#pragma unroll 4
  for (int c = 0; c < (NSTATE * NSTATE) / (NTHREADS * 4); ++c) {
    const int k = c * (NTHREADS * 4) + j * 4;
    async_load_b128(lt + k, tr + k);
  }
#else
  for (int c = 0; c < (NSTATE * NSTATE) / NTHREADS; ++c) {
    const int k = c * NTHREADS + j;
    tr[k] = lt[k];
  }
#endif

  // delta0 = emissions[:,0]
  dA[j] = emG[j];

#if HAVE_ASYNC_LDS
  // Prefetch emissions for t=1 into buffer parity 1, then drain the trans
  // loads (async completes in order: waiting <=1 leaves only em[1] pending).
  async_load_b32(emG + NSTATE + j, emb + NSTATE + j);
  wait_async_le1();
#endif
  __syncthreads();   // trans + delta0 visible to all waves

  float* cur = dA;
  float* nxt = dB;

  for (int tt = 1; tt < TLEN; ++tt) {
#if HAVE_ASYNC_LDS
    if (tt + 1 < TLEN)
      async_load_b32(emG + (size_t)(tt + 1) * NSTATE + j,
                     emb + ((tt + 1) & 1) * NSTATE + j);
#else
    if (tt + 1 < TLEN)
      __builtin_prefetch(emG + (size_t)(tt + 1) * NSTATE + j, 0, 0);
#endif

    // Max-plus reduction over previous state i with NCHAIN independent
    // accumulator chains (breaks the serial cmp/cndmask dependence so the
    // VALU stays issue-bound with only 2 waves/SIMD). delta chunk loads are
    // broadcast ds_load_b128; trans reads are lane-consecutive (bank-conflict
    // free). Within a chain, ascending i + strict '>' keeps the FIRST max.
    float bestv[NCHAIN];
    int   besti[NCHAIN];
#pragma unroll
    for (int q = 0; q < NCHAIN; ++q) { bestv[q] = -3.402823466e38f; besti[q] = 0; }

#pragma unroll 2
    for (int i = 0; i < NSTATE; i += NCHAIN) {
      const float4 dlo = *(const float4*)(cur + i);
      const float4 dhi = *(const float4*)(cur + i + 4);
      float s;
      s = dlo.x + tr[(i + 0) * NSTATE + j];
      if (s > bestv[0]) { bestv[0] = s; besti[0] = i + 0; }
      s = dlo.y + tr[(i + 1) * NSTATE + j];
      if (s > bestv[1]) { bestv[1] = s; besti[1] = i + 1; }
      s = dlo.z + tr[(i + 2) * NSTATE + j];
      if (s > bestv[2]) { bestv[2] = s; besti[2] = i + 2; }
      s = dlo.w + tr[(i + 3) * NSTATE + j];
      if (s > bestv[3]) { bestv[3] = s; besti[3] = i + 3; }
      s = dhi.x + tr[(i + 4) * NSTATE + j];
      if (s > bestv[4]) { bestv[4] = s; besti[4] = i + 4; }
      s = dhi.y + tr[(i + 5) * NSTATE + j];
      if (s > bestv[5]) { bestv[5] = s; besti[5] = i + 5; }
      s = dhi.z + tr[(i + 6) * NSTATE + j];
      if (s > bestv[6]) { bestv[6] = s; besti[6] = i + 6; }
      s = dhi.w + tr[(i + 7) * NSTATE + j];
      if (s > bestv[7]) { bestv[7] = s; besti[7] = i + 7; }
    }

    // Cross-chain merge with exact jnp.argmax tie rule: larger value wins;
    // equal value -> smaller index wins (global first-max).
    float best = bestv[0];
    int   amax = besti[0];
#pragma unroll
    for (int q = 1; q < NCHAIN; ++q) {
      if (bestv[q] > best || (bestv[q] == best && besti[q] < amax)) {
        best = bestv[q];
        amax = besti[q];
      }
    }

    // Emission for time tt.
#if HAVE_ASYNC_LDS
    if (tt + 1 < TLEN) wait_async_le1();  // em[tt] done (in-order), em[tt+1] may fly
    else               wait_async_le0();
    const float e = emb[(tt & 1) * NSTATE + j];
#else
    const float e = emG[(size_t)tt * NSTATE + j];
#endif

    nxt[j] = best + e;
    bp[(size_t)(tt - 1) * (BATCH * NSTATE) + (size_t)b * NSTATE + j] =
        (uint8_t)amax;

    __syncthreads();                 // nxt fully written before it becomes cur
    float* t = cur; cur = nxt; nxt = t;
  }

  // ---- Final argmax over delta_{T-1} (first-max tie rule) ------------------
  if (j == 0) {
    float bs = cur[0];
    int   as_ = 0;
    for (int k = 1; k < NSTATE; ++k) {
      if (cur[k] > bs) { bs = cur[k]; as_ = k; }
    }
    out_base[(size_t)BATCH * TLEN + b] = bs;  // best_score tail of d_out
    last_state[b] = as_;
  }
}

// ---------------------------------------------------------------------------
// Backtrack: 64 independent dependent-load chains (one per batch), all in
// flight at once; backpointers are L2-resident (33.5 MB << 192 MB L2).
// ---------------------------------------------------------------------------
__global__ __launch_bounds__(BATCH, 1)
void viterbi_backtrack_kernel(const uint8_t* __restrict__ bp,
                              const int* __restrict__ last_state,
                              int* __restrict__ path)   // [B,T] int32
{
  const int b = threadIdx.x;
  if (b >= BATCH) return;
  int s = last_state[b];
  path[(size_t)b * TLEN + (TLEN - 1)] = s;
  for (int t = TLEN - 2; t >= 0; --t) {
    s = (int)bp[(size_t)t * (BATCH * NSTATE) + (size_t)b * NSTATE + s];
    path[(size_t)b * TLEN + t] = s;
  }
}

// ---------------------------------------------------------------------------
// Host launcher
// ---------------------------------------------------------------------------
extern "C" void kernel_launch(void* const* d_in, const int* in_sizes, int n_in,
                              void* d_out, int out_size, void* d_ws, size_t ws_size,
                              hipStream_t stream) {
  (void)in_sizes; (void)n_in; (void)out_size; (void)ws_size;

  const float* em = (const float*)d_in[0];   // [B,T,S] f32
  const float* lt = (const float*)d_in[1];   // [S,S]   f32

  uint8_t* bp = (uint8_t*)d_ws;                                   // 33.5 MB
  int* last_state = (int*)(bp + (size_t)(TLEN - 1) * BATCH * NSTATE);

  int*   path  = (int*)d_out;                 // first B*T elems (int32 bits)
  float* fbase = (float*)d_out;               // scores at [B*T .. B*T+B)

  // LDS: 256KB trans + 2x1KB delta + 2x1KB emission buffers = 266240 B.
  // Requires CDNA5's 320KB-per-workgroup LDS.
  const size_t smem_bytes =
      (size_t)(NSTATE * NSTATE + 2 * NSTATE + 2 * NSTATE) * sizeof(float);

  hipLaunchKernelGGL(viterbi_fwd_kernel, dim3(BATCH), dim3(NTHREADS),
                     smem_bytes, stream, em, lt, bp, last_state, fbase);

  hipLaunchKernelGGL(viterbi_backtrack_kernel, dim3(1), dim3(BATCH),
                     0, stream, bp, last_state, path);
}